// GaussianConvBlur_51823075394123
// MI455X (gfx1250) — compile-verified
//
#include <hip/hip_runtime.h>
#include <hip/hip_bf16.h>

typedef __attribute__((ext_vector_type(2))) float v2f;
typedef __attribute__((ext_vector_type(8))) float v8f;

#define IMG 256
#define N_IMAGES (16 * 64)
#define ATAB_N (7 * 6 * 32 * 2)   // [j][q][lane][v] f32 entries

// ---------------------------------------------------------------------------
// Prep: build effective kernel and the 7 band matrices (A_j) in WMMA A-layout.
// A_j[m][k] = keff[k-m][j] for 0 <= k-m < 7, k in [0,22), else 0.
// Lane layout (f32 16x16x4 A): lane<16 -> m=lane, k=4q+{0,1}(v);
//                              lane>=16 -> m=lane-16, k=4q+{2,3}(v).
// ---------------------------------------------------------------------------
__global__ void __launch_bounds__(128)
gauss_prep(const float* __restrict__ w, const float* __restrict__ noise,
           const float* __restrict__ eps, float* __restrict__ atab_g) {
  __shared__ float keff[49];
  const int t = threadIdx.x;
  if (t < 49) {
    float m = 0.f;
#pragma unroll
    for (int i = 0; i < 49; ++i) m += eps[i];
    m *= (1.0f / 49.0f);
    // weight[0][0][dy][dx] are the first 49 floats of the (64,64,7,7) tensor
    keff[t] = w[t] + noise[0] * (eps[t] - m);
  }
  __syncthreads();
  for (int e = t; e < ATAB_N; e += 128) {
    const int v    = e & 1;
    const int lane = (e >> 1) & 31;
    const int jq   = e >> 6;          // j*6 + q
    const int q    = jq % 6;
    const int j    = jq / 6;
    const int m_   = lane & 15;
    const int k    = 4 * q + ((lane >> 4) << 1) + v;
    const int dy   = k - m_;
    atab_g[e] = (dy >= 0 && dy < 7 && k < 22) ? keff[dy * 7 + j] : 0.f;
  }
}

// ---------------------------------------------------------------------------
// Main: each 128-thread block (4 waves) computes a 32x32 output region of one
// image. LDS holds a zero-padded 40x40 input patch + the A-band table.
// Each wave does one 16x16 quadrant via 42x V_WMMA_F32_16X16X4_F32.
// ---------------------------------------------------------------------------
__global__ void __launch_bounds__(128)
gauss_conv(const float* __restrict__ x, const float* __restrict__ atab_g,
           float* __restrict__ out) {
  __shared__ float patch[40 * 40];
  __shared__ float atab[ATAB_N];

  const int tid = threadIdx.x;
  const int img = blockIdx.y;                 // 0..1023 (batch*channel)
  const int rh  = (blockIdx.x >> 3) << 5;     // region row origin
  const int rw  = (blockIdx.x & 7) << 5;      // region col origin
  const float* __restrict__ xim = x + (size_t)img * (IMG * IMG);

  for (int e = tid; e < ATAB_N; e += 128) atab[e] = atab_g[e];

  // Zero-padded 40x40 patch: rows/cols [-3, 36] of the 32x32 region.
  for (int idx = tid; idx < 1600; idx += 128) {
    const int pr = idx / 40;
    const int pc = idx - pr * 40;
    const int h = rh + pr - 3;
    const int w = rw + pc - 3;
    float v = 0.f;
    if ((unsigned)h < IMG && (unsigned)w < IMG) v = xim[h * IMG + w];
    patch[idx] = v;
  }
  __syncthreads();

  const int wave = tid >> 5;                  // quadrant per wave
  const int lane = tid & 31;
  const int th = (wave >> 1) << 4;
  const int tw = (wave & 1) << 4;
  const int hi = lane >> 4;                   // half-wave select
  const int n  = lane & 15;                   // N column
  const int col0 = tw + n;

  v8f acc0 = {};
  v8f acc1 = {};
#pragma unroll
  for (int q = 0; q < 6; ++q) {
    // B (4x16) layout: lane<16 rows 4q+{0,1}, lane>=16 rows 4q+{2,3}
    const int kr = th + 4 * q + hi * 2;
    const float* __restrict__ prow = &patch[kr * 40 + col0];
#pragma unroll
    for (int j = 0; j < 7; ++j) {
      const int ai = ((j * 6 + q) * 32 + lane) * 2;
      v2f A = {atab[ai], atab[ai + 1]};
      v2f B = {prow[j], prow[40 + j]};
      if (q & 1)
        acc1 = __builtin_amdgcn_wmma_f32_16x16x4_f32(
            false, A, false, B, (short)0, acc1, false, false);
      else
        acc0 = __builtin_amdgcn_wmma_f32_16x16x4_f32(
            false, A, false, B, (short)0, acc0, false, false);
    }
  }

  // D layout: lane n = N, VGPR r -> M = r + 8*hi
  float* __restrict__ o =
      out + (size_t)img * (IMG * IMG) + (size_t)(rh + th) * IMG + (rw + tw);
#pragma unroll
  for (int r = 0; r < 8; ++r) {
    o[(r + hi * 8) * IMG + n] = acc0[r] + acc1[r];
  }
}

extern "C" void kernel_launch(void* const* d_in, const int* in_sizes, int n_in,
                              void* d_out, int out_size, void* d_ws, size_t ws_size,
                              hipStream_t stream) {
  const float* x      = (const float*)d_in[0];
  const float* weight = (const float*)d_in[1];
  const float* noise  = (const float*)d_in[2];
  const float* eps    = (const float*)d_in[3];
  float* out  = (float*)d_out;
  float* atab = (float*)d_ws;   // needs ATAB_N*4 = 10752 bytes

  gauss_prep<<<1, 128, 0, stream>>>(weight, noise, eps, atab);

  dim3 grid(64, N_IMAGES);      // 8x8 regions of 32x32 per image, 1024 images
  gauss_conv<<<grid, 128, 0, stream>>>(x, atab, out);
}